// Aggregator_67577015436449
// MI455X (gfx1250) — compile-verified
//
#include <hip/hip_runtime.h>
#include <hip/hip_bf16.h>

typedef float v2f __attribute__((ext_vector_type(2)));
typedef float v8f __attribute__((ext_vector_type(8)));

#define D 128
#define LEAKY 0.01f

// ---------------------------------------------------------------------------
// Kernel 1: zero the N_h accumulator region in workspace
// ---------------------------------------------------------------------------
__global__ __launch_bounds__(256) void kagat_zero(float* __restrict__ nh, int n_elems) {
    int tid = blockIdx.x * blockDim.x + threadIdx.x;
    int stride = gridDim.x * blockDim.x;
    for (int i = tid; i < n_elems; i += stride) nh[i] = 0.0f;
}

// ---------------------------------------------------------------------------
// Kernel 2: transpose W1, W2 (row-major [j,k]) into Wt (row-major [k,j])
// so GEMM B-fragment loads are contiguous across lanes.
// ---------------------------------------------------------------------------
__global__ __launch_bounds__(256) void kagat_transpose(const float* __restrict__ W1,
                                                       const float* __restrict__ W2,
                                                       float* __restrict__ Wt1,
                                                       float* __restrict__ Wt2) {
    int tid = blockIdx.x * blockDim.x + threadIdx.x;  // 0 .. 128*128-1
    if (tid >= D * D) return;
    int k = tid >> 7;        // row of Wt
    int n = tid & (D - 1);   // col of Wt
    Wt1[tid] = W1[n * D + k];
    Wt2[tid] = W2[n * D + k];
}

// ---------------------------------------------------------------------------
// Kernel 3: edge scatter.  One wave32 per edge; lane l owns cols [4l, 4l+4).
// msg = entity[src] * att[e], atomically summed into nh[dst].
// ---------------------------------------------------------------------------
__global__ __launch_bounds__(256) void kagat_scatter(const float* __restrict__ entity,
                                                     const float* __restrict__ att,
                                                     const int* __restrict__ src_idx,
                                                     const int* __restrict__ dst_idx,
                                                     float* __restrict__ nh,
                                                     int n_edges) {
    int tid  = blockIdx.x * blockDim.x + threadIdx.x;
    int e    = tid >> 5;
    if (e >= n_edges) return;
    int lane = tid & 31;
    int col  = lane << 2;

    int   src = src_idx[e];
    int   dst = dst_idx[e];
    float a   = att[e];

    const float4 v = *(const float4*)(entity + (size_t)src * D + col);
    float* p = nh + (size_t)dst * D + col;
    unsafeAtomicAdd(p + 0, v.x * a);
    unsafeAtomicAdd(p + 1, v.y * a);
    unsafeAtomicAdd(p + 2, v.z * a);
    unsafeAtomicAdd(p + 3, v.w * a);
}

// ---------------------------------------------------------------------------
// Kernel 4: fused bi-interaction GEMM epilogue via V_WMMA_F32_16X16X4_F32.
//   out = lrelu((e+h)@W1^T + b1) + lrelu((e*h)@W2^T + b2)
// Block = 256 threads = 8 waves: 2 waves in M (32 rows) x 4 waves in N (128 cols).
// Each wave: 16 rows x 32 cols -> 2 N-tiles, 2 matrices -> 4 v8f accumulators.
// ---------------------------------------------------------------------------
__global__ __launch_bounds__(256) void kagat_gemm(const float* __restrict__ entity,
                                                  const float* __restrict__ nh,
                                                  const float* __restrict__ Wt1,
                                                  const float* __restrict__ Wt2,
                                                  const float* __restrict__ b1,
                                                  const float* __restrict__ b2,
                                                  float* __restrict__ out,
                                                  int n_nodes) {
    const int lane  = threadIdx.x & 31;
    const int wave  = threadIdx.x >> 5;
    const int mtile = blockIdx.x * 2 + (wave >> 2);  // 16-row tile index
    const int m0    = mtile * 16;
    const int n0    = (wave & 3) * 32;               // 32-col slab
    const int laneM = lane & 15;
    const int kl    = (lane >> 4) << 1;              // 0 or 2: K sub-offset

    if (m0 >= n_nodes) return;                       // whole-wave uniform exit

    // Clamp A-row so EXEC stays all-ones through the WMMA loop.
    int rowA = m0 + laneM;
    if (rowA >= n_nodes) rowA = n_nodes - 1;
    const float* erow = entity + (size_t)rowA * D;
    const float* hrow = nh     + (size_t)rowA * D;

    v8f acc1[2] = {v8f{0.f,0.f,0.f,0.f,0.f,0.f,0.f,0.f},
                   v8f{0.f,0.f,0.f,0.f,0.f,0.f,0.f,0.f}};
    v8f acc2[2] = {v8f{0.f,0.f,0.f,0.f,0.f,0.f,0.f,0.f},
                   v8f{0.f,0.f,0.f,0.f,0.f,0.f,0.f,0.f}};

    for (int k = 0; k < D; k += 4) {
        // A fragments (both matrices share the same loads)
        float e0 = erow[k + kl], e1 = erow[k + kl + 1];
        float h0 = hrow[k + kl], h1 = hrow[k + kl + 1];
        v2f a1 = {e0 + h0, e1 + h1};   // (entity + N_h)
        v2f a2 = {e0 * h0, e1 * h1};   // (entity * N_h)

#pragma unroll
        for (int t = 0; t < 2; ++t) {
            const int n = n0 + t * 16 + laneM;
            // B fragment: B[kk][n] = W^T[kk][n] = Wt[kk*D + n]
            v2f bf1 = {Wt1[(k + kl) * D + n], Wt1[(k + kl + 1) * D + n]};
            v2f bf2 = {Wt2[(k + kl) * D + n], Wt2[(k + kl + 1) * D + n]};
            acc1[t] = __builtin_amdgcn_wmma_f32_16x16x4_f32(
                false, a1, false, bf1, (short)0, acc1[t], false, false);
            acc2[t] = __builtin_amdgcn_wmma_f32_16x16x4_f32(
                false, a2, false, bf2, (short)0, acc2[t], false, false);
        }
    }

    // Epilogue: bias + leaky_relu + sum of both branches
    const int rbase = m0 + ((lane >> 4) << 3);       // lanes 16-31 hold rows +8
#pragma unroll
    for (int t = 0; t < 2; ++t) {
        const int col   = n0 + t * 16 + laneM;
        const float bb1 = b1[col];
        const float bb2 = b2[col];
#pragma unroll
        for (int v = 0; v < 8; ++v) {
            const int row = rbase + v;
            if (row < n_nodes) {
                float o1 = acc1[t][v] + bb1;
                o1 = (o1 > 0.f) ? o1 : o1 * LEAKY;
                float o2 = acc2[t][v] + bb2;
                o2 = (o2 > 0.f) ? o2 : o2 * LEAKY;
                out[(size_t)row * D + col] = o1 + o2;
            }
        }
    }
}

// ---------------------------------------------------------------------------
// Launcher
// ---------------------------------------------------------------------------
extern "C" void kernel_launch(void* const* d_in, const int* in_sizes, int n_in,
                              void* d_out, int out_size, void* d_ws, size_t ws_size,
                              hipStream_t stream) {
    const float* entity = (const float*)d_in[0];
    const float* att    = (const float*)d_in[1];
    const int*   srcI   = (const int*)d_in[2];
    const int*   dstI   = (const int*)d_in[3];
    const float* W1     = (const float*)d_in[4];
    const float* b1     = (const float*)d_in[5];
    const float* W2     = (const float*)d_in[6];
    const float* b2     = (const float*)d_in[7];
    float*       out    = (float*)d_out;

    const int n_nodes = in_sizes[0] / D;
    const int n_edges = in_sizes[1];

    // Workspace layout: [N_h (n_nodes*D f32)] [Wt1 (D*D)] [Wt2 (D*D)]
    float* nh  = (float*)d_ws;
    float* Wt1 = nh + (size_t)n_nodes * D;
    float* Wt2 = Wt1 + (size_t)D * D;

    // 1) zero N_h
    {
        int elems  = n_nodes * D;
        int blocks = (elems + 256 * 8 - 1) / (256 * 8);
        kagat_zero<<<blocks, 256, 0, stream>>>(nh, elems);
    }
    // 2) transpose weights
    kagat_transpose<<<(D * D + 255) / 256, 256, 0, stream>>>(W1, W2, Wt1, Wt2);
    // 3) edge scatter: one wave per edge
    {
        long long threads = (long long)n_edges * 32;
        int blocks = (int)((threads + 255) / 256);
        kagat_scatter<<<blocks, 256, 0, stream>>>(entity, att, srcI, dstI, nh, n_edges);
    }
    // 4) fused WMMA GEMM + activation
    {
        int mtiles = (n_nodes + 15) / 16;
        int blocks = (mtiles + 1) / 2;   // 2 M-tiles per block
        kagat_gemm<<<blocks, 256, 0, stream>>>(entity, nh, Wt1, Wt2, b1, b2, out, n_nodes);
    }
}